// GCN_10024453669129
// MI455X (gfx1250) — compile-verified
//
#include <hip/hip_runtime.h>

// ---------------------------------------------------------------------------
// GCN 2-layer forward for MI455X (gfx1250, wave32).
// GEMMs use V_WMMA_F32_16X16X4_F32 (exact fp32 path; GEMMs are not the
// bottleneck -- edge scatter is, and it is L2-resident on a 192MB L2).
// ---------------------------------------------------------------------------

typedef __attribute__((ext_vector_type(2))) float v2f;
typedef __attribute__((ext_vector_type(8))) float v8f;

// ------------------------------ utility ------------------------------------

__global__ void GCN_zero_f32(float* __restrict__ p, long n) {
  long i = (long)blockIdx.x * blockDim.x + threadIdx.x;
  long stride = (long)gridDim.x * blockDim.x;
  for (; i < n; i += stride) p[i] = 0.0f;
}

__global__ void GCN_degree(const int* __restrict__ src, const int* __restrict__ dst,
                           float* __restrict__ deg_out, float* __restrict__ deg_in,
                           int ne) {
  int i = blockIdx.x * blockDim.x + threadIdx.x;
  if (i < ne) {
    atomicAdd(&deg_out[src[i]], 1.0f);
    atomicAdd(&deg_in[dst[i]], 1.0f);
  }
}

__global__ void GCN_norms(const float* __restrict__ deg_out, const float* __restrict__ deg_in,
                          float* __restrict__ nsrc, float* __restrict__ ndst, int n) {
  int i = blockIdx.x * blockDim.x + threadIdx.x;
  if (i < n) {
    nsrc[i] = rsqrtf(fmaxf(deg_out[i], 1.0f));
    ndst[i] = rsqrtf(fmaxf(deg_in[i], 1.0f));
  }
}

// ------------------------- WMMA f32 GEMM -----------------------------------
// Y[row, col] = sum_k ( X[row,k] * scale[row] ) * W[k, col]
// One wave per 16x16 output tile; block covers one 16-row stripe and all NT
// column tiles. X-tile (scaled) staged in LDS with +4 word padding so the
// strided A-fragment reads hit 16 distinct banks. If NC is not a multiple of
// 16, W is staged in LDS zero-padded to NT*16 columns so the inner loop has
// NO per-lane conditional loads (avoids exec-mask branch serialization).
// Store epilogue has a hoisted full-tile fast path (unconditional 8-store
// clause) -- taken by every block when nrows % 16 == 0.
template <int K, int NT, int NC>
__global__ __launch_bounds__(32 * NT)
void GCN_gemm_wmma(const float* __restrict__ X, const float* __restrict__ scale,
                   const float* __restrict__ W, float* __restrict__ Y, int nrows) {
  constexpr int NCP = NT * 16;             // padded column count
  constexpr bool EXACT = (NC == NCP);      // true -> read W straight from global
  constexpr int PITCH = K + 4;
  __shared__ float sa[16 * PITCH];
  __shared__ float sw[EXACT ? 4 : K * NCP];

  const int rowbase = blockIdx.x * 16;

  // Cooperative coalesced staging: 16 rows x K cols of X, fused degree scaling.
  for (int idx = threadIdx.x; idx < 16 * (K / 4); idx += blockDim.x) {
    const int r = idx / (K / 4);
    const int c = idx - r * (K / 4);
    const int row = rowbase + r;
    float4 v = make_float4(0.f, 0.f, 0.f, 0.f);
    float s = 0.0f;
    if (row < nrows) {
      v = ((const float4*)(X + (long)row * K))[c];
      s = scale[row];
    }
    float* d = &sa[r * PITCH + 4 * c];
    d[0] = v.x * s; d[1] = v.y * s; d[2] = v.z * s; d[3] = v.w * s;
  }

  if constexpr (!EXACT) {
    // Stage W zero-padded to NCP columns (e.g. 128x48 = 24KB for layer 2).
    for (int idx = threadIdx.x; idx < K * NCP; idx += blockDim.x) {
      const int k = idx / NCP;
      const int c = idx - k * NCP;
      sw[idx] = (c < NC) ? W[k * NC + c] : 0.0f;
    }
  }
  __syncthreads();

  const int lane = threadIdx.x & 31;
  const int wave = threadIdx.x >> 5;
  const int colbase = wave * 16;
  const int half = lane >> 4;       // 0: lanes 0-15, 1: lanes 16-31
  const int l15 = lane & 15;
  const int koff = half * 2;        // A/B fragment: upper lanes hold K=2,3

  const float* aptr = &sa[l15 * PITCH + koff];
  const int bcol = colbase + l15;

  v8f acc = {};
#pragma unroll 8
  for (int k0 = 0; k0 < K; k0 += 4) {
    v2f a;
    a.x = aptr[k0];
    a.y = aptr[k0 + 1];
    v2f b;
    if constexpr (EXACT) {
      // Unconditional: each half-wave reads a contiguous 64B row segment.
      b.x = W[(k0 + koff) * NC + bcol];
      b.y = W[(k0 + koff + 1) * NC + bcol];
    } else {
      // Unconditional LDS reads from the zero-padded staged W.
      b.x = sw[(k0 + koff) * NCP + bcol];
      b.y = sw[(k0 + koff + 1) * NCP + bcol];
    }
    acc = __builtin_amdgcn_wmma_f32_16x16x4_f32(
        /*neg_a=*/false, a, /*neg_b=*/false, b,
        /*c_mod=*/(short)0, acc, /*reuse_a=*/false, /*reuse_b=*/false);
  }

  // C/D layout: VGPR r -> row (rowbase + half*8 + r), col = colbase + l15.
  if (EXACT || bcol < NC) {
    float* yp = Y + (long)(rowbase + half * 8) * NC + bcol;
    if (rowbase + 16 <= nrows) {
      // Fast path (always taken when nrows % 16 == 0): unconditional stores.
#pragma unroll
      for (int r = 0; r < 8; ++r) yp[(long)r * NC] = acc[r];
    } else {
      const int rbase = rowbase + half * 8;
#pragma unroll
      for (int r = 0; r < 8; ++r) {
        if (rbase + r < nrows) yp[(long)r * NC] = acc[r];
      }
    }
  }
}

// --------------------------- edge scatter ----------------------------------
// One wave per edge. 128 features: each lane loads float4 (512B coalesced
// gather per wave, L2-resident) and does 4 f32 atomic adds.
__global__ void GCN_scatter128(const float* __restrict__ H,
                               const int* __restrict__ src, const int* __restrict__ dst,
                               float* __restrict__ AGG, int ne) {
  const int wid = (int)((blockIdx.x * (long)blockDim.x + threadIdx.x) >> 5);
  const int lane = threadIdx.x & 31;
  if (wid >= ne) return;
  const int s = src[wid];
  const int d = dst[wid];
  const float4 v = ((const float4*)(H + (long)s * 128))[lane];
  float* out = AGG + (long)d * 128 + lane * 4;
  atomicAdd(out + 0, v.x);
  atomicAdd(out + 1, v.y);
  atomicAdd(out + 2, v.z);
  atomicAdd(out + 3, v.w);
}

// 40 features: lane f and (lanes 0-7) f+32.
__global__ void GCN_scatter40(const float* __restrict__ H,
                              const int* __restrict__ src, const int* __restrict__ dst,
                              float* __restrict__ AGG, int ne) {
  const int wid = (int)((blockIdx.x * (long)blockDim.x + threadIdx.x) >> 5);
  const int lane = threadIdx.x & 31;
  if (wid >= ne) return;
  const int s = src[wid];
  const int d = dst[wid];
  const float* hp = H + (long)s * 40;
  float* out = AGG + (long)d * 40;
  atomicAdd(out + lane, hp[lane]);
  if (lane < 8) atomicAdd(out + lane + 32, hp[lane + 32]);
}

// --------------------------- post passes -----------------------------------

__global__ void GCN_post1(const float* __restrict__ AGG, const float* __restrict__ ndst,
                          const float* __restrict__ b, float* __restrict__ H1, long total) {
  long idx = (long)blockIdx.x * blockDim.x + threadIdx.x;
  long stride = (long)gridDim.x * blockDim.x;
  for (; idx < total; idx += stride) {
    const int i = (int)(idx >> 7);       // /128
    const int f = (int)(idx & 127);
    const float v = AGG[idx] * ndst[i] + b[f];
    H1[idx] = v > 0.0f ? v : 0.0f;       // ReLU
  }
}

__global__ void GCN_post2(float* __restrict__ OUT, const float* __restrict__ ndst,
                          const float* __restrict__ b, long total) {
  long idx = (long)blockIdx.x * blockDim.x + threadIdx.x;
  long stride = (long)gridDim.x * blockDim.x;
  for (; idx < total; idx += stride) {
    const int i = (int)(idx / 40);
    const int f = (int)(idx - (long)i * 40);
    OUT[idx] = OUT[idx] * ndst[i] + b[f];
  }
}

// ------------------------------ launcher -----------------------------------

extern "C" void kernel_launch(void* const* d_in, const int* in_sizes, int n_in,
                              void* d_out, int out_size, void* d_ws, size_t ws_size,
                              hipStream_t stream) {
  const float* x   = (const float*)d_in[0];   // [N, 256]
  const float* W1  = (const float*)d_in[1];   // [256, 128]
  const float* b1  = (const float*)d_in[2];   // [128]
  const float* W2  = (const float*)d_in[3];   // [128, 40]
  const float* b2  = (const float*)d_in[4];   // [40]
  const int* esrc  = (const int*)d_in[5];     // [E]
  const int* edst  = (const int*)d_in[6];     // [E]
  float* out = (float*)d_out;                 // [N, 40]

  const int nn = in_sizes[0] / 256;           // 50000
  const int ne = in_sizes[5];                 // 800000

  // Workspace layout (floats)
  const long NPAD = 50048;                    // nodes rounded to 64
  float* ws       = (float*)d_ws;
  float* deg_out  = ws;
  float* deg_in   = ws + NPAD;
  float* nsrc     = ws + 2 * NPAD;
  float* ndst     = ws + 3 * NPAD;
  float* h        = ws + 4 * NPAD;                       // [N,128] gemm1 out, then H1
  float* agg1     = ws + 4 * NPAD + (long)nn * 128;      // [N,128]
  float* h2       = ws + 4 * NPAD + 2L * (long)nn * 128; // [N,40]

  // 1) zero accumulators (must happen every call: graph replays)
  GCN_zero_f32<<<512, 256, 0, stream>>>(deg_out, 2 * NPAD);
  GCN_zero_f32<<<4096, 256, 0, stream>>>(agg1, (long)nn * 128);
  GCN_zero_f32<<<2048, 256, 0, stream>>>(out, (long)nn * 40);

  // 2) degrees + norms
  GCN_degree<<<(ne + 255) / 256, 256, 0, stream>>>(esrc, edst, deg_out, deg_in, ne);
  GCN_norms<<<(nn + 255) / 256, 256, 0, stream>>>(deg_out, deg_in, nsrc, ndst, nn);

  const int mtiles = (nn + 15) / 16;          // 3125

  // 3) layer 1: h = (x * nsrc) @ W1   [WMMA f32, 8 waves = 8 col tiles]
  GCN_gemm_wmma<256, 8, 128><<<mtiles, 256, 0, stream>>>(x, nsrc, W1, h, nn);

  // 4) scatter-add over edges (128 feats)
  GCN_scatter128<<<(ne + 7) / 8, 256, 0, stream>>>(h, esrc, edst, agg1, ne);

  // 5) h1 = relu(agg1 * ndst + b1)  (reuses h buffer)
  GCN_post1<<<8192, 256, 0, stream>>>(agg1, ndst, b1, h, (long)nn * 128);

  // 6) layer 2: h2 = (h1 * nsrc) @ W2  [WMMA f32, W2 staged zero-padded in LDS]
  GCN_gemm_wmma<128, 3, 40><<<mtiles, 96, 0, stream>>>(h, nsrc, W2, h2, nn);

  // 7) scatter-add (40 feats) directly into d_out
  GCN_scatter40<<<(ne + 7) / 8, 256, 0, stream>>>(h2, esrc, edst, out, ne);

  // 8) out = out * ndst + b2
  GCN_post2<<<4096, 256, 0, stream>>>(out, ndst, b2, (long)nn * 40);
}